// Attn_24352464569397
// MI455X (gfx1250) — compile-verified
//
#include <hip/hip_runtime.h>
#include <hip/hip_bf16.h>
#include <math.h>

typedef float v2f __attribute__((ext_vector_type(2)));
typedef float v4f __attribute__((ext_vector_type(4)));
typedef float v8f __attribute__((ext_vector_type(8)));

#define B_DIM 32
#define L_DIM 4096
#define H_DIM 1024

#define WAVES_PER_BLOCK 8
#define CHUNK 32                 // h-floats staged per step (1 cacheline per row)
#define LDS_STRIDE 36            // padded row stride: conflict-free ds_load_b64,
                                 // 144B keeps b128 stores 16B-aligned

// Phase 1: scores[b, l] = sum_h hidden[b,h] * enc[l,b,h]
// One wave per (b, 16-l tile). Coalesced global_load_b128 staging into LDS,
// WMMA A-operands via conflict-free ds_load_b64, B = hidden broadcast into
// all 16 columns. Accumulate with V_WMMA_F32_16X16X4_F32.
__global__ __launch_bounds__(256) void attn_scores_wmma(
    const float* __restrict__ hidden,
    const float* __restrict__ enc,
    float* __restrict__ scores)
{
    __shared__ float smem[WAVES_PER_BLOCK * 16 * LDS_STRIDE];

    const int tid  = threadIdx.x;
    const int lane = tid & 31;
    const int wid  = tid >> 5;
    const int wave = blockIdx.x * WAVES_PER_BLOCK + wid;

    const int tiles_per_b = L_DIM / 16;          // 256
    const int b  = wave / tiles_per_b;
    const int l0 = (wave % tiles_per_b) * 16;
    if (b >= B_DIM) return;                      // wave-uniform; EXEC stays full

    // WMMA operand roles
    const int m     = lane & 15;                 // A row (l offset), D column
    const int khalf = lane >> 4;                 // 0: K=0,1   1: K=2,3

    // Staging roles: 8 contiguous lanes cover one 128B row-line
    const int srow = lane >> 3;                  // 0..3 (rows 4i+srow)
    const int scol = (lane & 7) * 4;             // 0,4,...,28

    float* lds = smem + wid * (16 * LDS_STRIDE);

    const size_t RS = (size_t)B_DIM * H_DIM;     // enc row stride over l
    const float* __restrict__ gp =
        enc + (size_t)(l0 + srow) * RS + (size_t)b * H_DIM + scol;
    const float* __restrict__ hrow =
        hidden + (size_t)b * H_DIM + 2 * khalf;
    const float* __restrict__ ap = lds + m * LDS_STRIDE + 2 * khalf;
    float* __restrict__ sp = lds + srow * LDS_STRIDE + scol;

    v8f acc0 = {};
    v8f acc1 = {};
    v8f acc2 = {};
    v8f acc3 = {};

    // Prefetch chunk 0 (4 x b128, each covering 4 full cachelines wave-wide)
    v4f g0 = *(const v4f*)(gp + 0 * 4 * RS);
    v4f g1 = *(const v4f*)(gp + 1 * 4 * RS);
    v4f g2 = *(const v4f*)(gp + 2 * 4 * RS);
    v4f g3 = *(const v4f*)(gp + 3 * 4 * RS);

    const int NCHUNK = H_DIM / CHUNK;            // 32
    for (int c = 0; c < NCHUNK; ++c) {
        // Stage current chunk into LDS (ds in-order per wave: reads of the
        // previous chunk issued earlier are serviced first).
        *(v4f*)(sp + 0 * 4 * LDS_STRIDE) = g0;
        *(v4f*)(sp + 1 * 4 * LDS_STRIDE) = g1;
        *(v4f*)(sp + 2 * 4 * LDS_STRIDE) = g2;
        *(v4f*)(sp + 3 * 4 * LDS_STRIDE) = g3;

        // Prefetch next chunk while WMMAs below execute.
        if (c + 1 < NCHUNK) {
            const float* gq = gp + (c + 1) * CHUNK;
            g0 = *(const v4f*)(gq + 0 * 4 * RS);
            g1 = *(const v4f*)(gq + 1 * 4 * RS);
            g2 = *(const v4f*)(gq + 2 * 4 * RS);
            g3 = *(const v4f*)(gq + 3 * 4 * RS);
        }

        // 8 K-steps of 4 over this chunk; A from LDS (conflict-free b64),
        // B from hidden (uniform-address L0 hits, broadcast over N).
        const float* hp = hrow + c * CHUNK;
        #pragma unroll
        for (int kk = 0; kk < 8; kk += 4) {
            v2f a0 = *(const v2f*)(ap + (kk + 0) * 4);
            v2f a1 = *(const v2f*)(ap + (kk + 1) * 4);
            v2f a2 = *(const v2f*)(ap + (kk + 2) * 4);
            v2f a3 = *(const v2f*)(ap + (kk + 3) * 4);
            v2f b0 = *(const v2f*)(hp + (kk + 0) * 4);
            v2f b1 = *(const v2f*)(hp + (kk + 1) * 4);
            v2f b2 = *(const v2f*)(hp + (kk + 2) * 4);
            v2f b3 = *(const v2f*)(hp + (kk + 3) * 4);
            acc0 = __builtin_amdgcn_wmma_f32_16x16x4_f32(false, a0, false, b0,
                                                         (short)0, acc0, false, false);
            acc1 = __builtin_amdgcn_wmma_f32_16x16x4_f32(false, a1, false, b1,
                                                         (short)0, acc1, false, false);
            acc2 = __builtin_amdgcn_wmma_f32_16x16x4_f32(false, a2, false, b2,
                                                         (short)0, acc2, false, false);
            acc3 = __builtin_amdgcn_wmma_f32_16x16x4_f32(false, a3, false, b3,
                                                         (short)0, acc3, false, false);
        }
    }
    v8f acc = (acc0 + acc1) + (acc2 + acc3);

    // C/D layout: lanes 0-15 hold M=0..7 in V0..V7 (col = lane),
    // lanes 16-31 hold M=8..15. All columns identical -> col 0 of each half.
    if (m == 0) {
        float* p = scores + (size_t)b * L_DIM + l0 + khalf * 8;
        v4f lo = {acc[0], acc[1], acc[2], acc[3]};
        v4f hi = {acc[4], acc[5], acc[6], acc[7]};
        *(v4f*)(p)     = lo;
        *(v4f*)(p + 4) = hi;
    }
}

// Phase 2: in-place softmax over L for each b. One block per b.
__global__ __launch_bounds__(1024) void attn_softmax(float* __restrict__ data)
{
    const int b   = blockIdx.x;
    const int tid = threadIdx.x;
    float* row = data + (size_t)b * L_DIM;

    __shared__ float red[32];
    __shared__ float s_max, s_sum;

    float x[4];
    float lm = -INFINITY;
    #pragma unroll
    for (int i = 0; i < 4; ++i) {
        x[i] = row[tid + i * 1024];
        lm = fmaxf(lm, x[i]);
    }
    #pragma unroll
    for (int off = 16; off; off >>= 1)
        lm = fmaxf(lm, __shfl_xor(lm, off, 32));
    if ((tid & 31) == 0) red[tid >> 5] = lm;
    __syncthreads();
    if (tid < 32) {
        float v = red[tid];
        #pragma unroll
        for (int off = 16; off; off >>= 1)
            v = fmaxf(v, __shfl_xor(v, off, 32));
        if (tid == 0) s_max = v;
    }
    __syncthreads();

    const float mx = s_max;
    float ls = 0.0f;
    #pragma unroll
    for (int i = 0; i < 4; ++i) {
        x[i] = expf(x[i] - mx);
        ls += x[i];
    }
    #pragma unroll
    for (int off = 16; off; off >>= 1)
        ls += __shfl_xor(ls, off, 32);
    if ((tid & 31) == 0) red[tid >> 5] = ls;
    __syncthreads();
    if (tid < 32) {
        float v = red[tid];
        #pragma unroll
        for (int off = 16; off; off >>= 1)
            v += __shfl_xor(v, off, 32);
        if (tid == 0) s_sum = v;
    }
    __syncthreads();

    const float inv = 1.0f / s_sum;
    #pragma unroll
    for (int i = 0; i < 4; ++i)
        row[tid + i * 1024] = x[i] * inv;
}

extern "C" void kernel_launch(void* const* d_in, const int* in_sizes, int n_in,
                              void* d_out, int out_size, void* d_ws, size_t ws_size,
                              hipStream_t stream) {
    (void)in_sizes; (void)n_in; (void)out_size; (void)d_ws; (void)ws_size;
    const float* hidden = (const float*)d_in[0];   // [1, 32, 1024] fp32
    const float* enc    = (const float*)d_in[1];   // [4096, 32, 1024] fp32
    float* out = (float*)d_out;                    // [32, 1, 4096] fp32

    const int total_waves = B_DIM * (L_DIM / 16);            // 8192
    const int blocks      = total_waves / WAVES_PER_BLOCK;   // 1024

    attn_scores_wmma<<<blocks, WAVES_PER_BLOCK * 32, 0, stream>>>(hidden, enc, out);
    attn_softmax<<<B_DIM, 1024, 0, stream>>>(out);
}